// GraphTransformer_19524921327916
// MI455X (gfx1250) — compile-verified
//
#include <hip/hip_runtime.h>
#include <hip/hip_bf16.h>

#define Nn   20000
#define Ee   320000
#define HIDc 128
#define HEADSc 8
#define Cc   16
#define Lc   3
#define Gc   64
#define OUTc 10

typedef __attribute__((ext_vector_type(2))) float v2f;
typedef __attribute__((ext_vector_type(8))) float v8f;

// ---------- helpers: order-preserving float <-> uint map for atomicMax ----------
__device__ __forceinline__ unsigned f2ord(float f) {
    unsigned u = __float_as_uint(f);
    return (u & 0x80000000u) ? ~u : (u | 0x80000000u);
}
__device__ __forceinline__ float ord2f(unsigned u) {
    return __uint_as_float((u & 0x80000000u) ? (u & 0x7fffffffu) : ~u);
}

// =====================================================================
// Y[M x 128] = act(X[M x 128] @ W[128 x 128] + b)   (row-major, M % 16 == 0)
// One 16-row block per workgroup; 8 waves, wave w owns 16x16 col tile w.
// Uses exact-fp32 V_WMMA_F32_16X16X4_F32, K-loop of 32 steps.
// =====================================================================
__global__ void gemm128_wmma(const float* __restrict__ X,
                             const float* __restrict__ W,
                             const float* __restrict__ bias,
                             float* __restrict__ Y,
                             int relu) {
    const int row0 = blockIdx.x << 4;
    const int wave = threadIdx.x >> 5;     // 0..7 : column tile
    const int lane = threadIdx.x & 31;
    const int col0 = wave << 4;
    const int ln   = lane & 15;
    const int hi   = lane >> 4;            // 0 or 1 (half-wave)
    const int arow = row0 + ln;            // A row held by this lane

    v8f acc = {};
    const float* xrow = X + (size_t)arow * HIDc;

    for (int k = 0; k < 128; k += 4) {
        // A (16x4): lanes 0-15 hold K=k..k+1, lanes 16-31 hold K=k+2..k+3
        const float2 av = *(const float2*)(xrow + k + hi * 2);
        v2f a; a.x = av.x; a.y = av.y;
        // B (4x16): same half-wave K split, N = col0+ln
        const float* bp = W + (size_t)(k + hi * 2) * HIDc + col0 + ln;
        v2f b; b.x = bp[0]; b.y = bp[HIDc];
        acc = __builtin_amdgcn_wmma_f32_16x16x4_f32(
            /*neg_a=*/false, a, /*neg_b=*/false, b,
            /*c_mod=*/(short)0, acc, /*reuse_a=*/false, /*reuse_b=*/false);
    }

    const float bv = bias[col0 + ln];
#pragma unroll
    for (int r = 0; r < 8; ++r) {
        // C/D layout: VGPR r -> M = r (+8 for upper half-wave), N = ln
        const int row = row0 + r + (hi << 3);
        float val = acc[r] + bv;
        if (relu) val = fmaxf(val, 0.0f);
        Y[(size_t)row * HIDc + col0 + ln] = val;
    }
}

// ---------- zero fill ----------
__global__ void fill_zero_u32(unsigned* __restrict__ p, int n) {
    int t = blockIdx.x * blockDim.x + threadIdx.x;
    if (t < n) p[t] = 0u;
}

// ---------- per-edge attention logits + segment max ----------
__global__ void edge_logits(const float* __restrict__ Q,
                            const float* __restrict__ Kmat,
                            const int* __restrict__ ei,
                            float* __restrict__ logits,
                            unsigned* __restrict__ nodemax) {
    int t = blockIdx.x * blockDim.x + threadIdx.x;
    if (t >= Ee * HEADSc) return;
    const int e = t >> 3, h = t & 7;
    const int src = ei[e], dst = ei[Ee + e];
    const float4* q = (const float4*)(Q + (size_t)dst * HIDc + h * Cc);
    const float4* k = (const float4*)(Kmat + (size_t)src * HIDc + h * Cc);
    float s = 0.0f;
#pragma unroll
    for (int i = 0; i < 4; ++i) {
        const float4 a = q[i], b = k[i];
        s += a.x * b.x + a.y * b.y + a.z * b.z + a.w * b.w;
    }
    s *= 0.25f;                       // 1/sqrt(C), C=16
    logits[t] = s;
    atomicMax(nodemax + (size_t)dst * HEADSc + h, f2ord(s));
}

// ---------- exp(logit - max) + segment sum ----------
__global__ void edge_exp(float* __restrict__ logits,
                         const unsigned* __restrict__ nodemax,
                         float* __restrict__ denom,
                         const int* __restrict__ ei) {
    int t = blockIdx.x * blockDim.x + threadIdx.x;
    if (t >= Ee * HEADSc) return;
    const int e = t >> 3, h = t & 7;
    const int dst = ei[Ee + e];
    const float m  = ord2f(nodemax[(size_t)dst * HEADSc + h]);
    const float ex = __expf(logits[t] - m);
    logits[t] = ex;
    atomicAdd(denom + (size_t)dst * HEADSc + h, ex);
}

// ---------- scatter alpha * V[src] into out[dst] (out pre-loaded with skip) ----------
// 32 lanes per edge, 4 channels per lane (channels 4-aligned => single head per lane)
__global__ void edge_scatter(const float* __restrict__ logits,
                             const float* __restrict__ denom,
                             const float* __restrict__ V,
                             const int* __restrict__ ei,
                             float* __restrict__ out) {
    int t = blockIdx.x * blockDim.x + threadIdx.x;
    if (t >= Ee * 32) return;
    const int e    = t >> 5;
    const int lane = t & 31;
    const int c0   = lane << 2;       // 0..124
    const int h    = c0 >> 4;         // head id
    const int src = ei[e], dst = ei[Ee + e];
    const float coef = logits[(size_t)e * HEADSc + h] /
                       denom[(size_t)dst * HEADSc + h];
    const float4 v = *(const float4*)(V + (size_t)src * HIDc + c0);
    float* o = out + (size_t)dst * HIDc + c0;
    atomicAdd(o + 0, v.x * coef);
    atomicAdd(o + 1, v.y * coef);
    atomicAdd(o + 2, v.z * coef);
    atomicAdd(o + 3, v.w * coef);
}

// ---------- in-place ReLU ----------
__global__ void relu_ip(float* __restrict__ p, int n) {
    int t = blockIdx.x * blockDim.x + threadIdx.x;
    if (t < n) p[t] = fmaxf(p[t], 0.0f);
}

// ---------- global add pool ----------
__global__ void pool_kernel(const float* __restrict__ x,
                            const int* __restrict__ batch,
                            float* __restrict__ pool) {
    int t = blockIdx.x * blockDim.x + threadIdx.x;
    if (t >= Nn * HIDc) return;
    const int n = t >> 7, c = t & 127;
    atomicAdd(pool + (size_t)batch[n] * HIDc + c, x[t]);
}

// ---------- classifier: out[G x 10] = h @ Wc + bc ----------
__global__ void classifier(const float* __restrict__ h,
                           const float* __restrict__ Wc,
                           const float* __restrict__ bc,
                           float* __restrict__ out) {
    int t = blockIdx.x * blockDim.x + threadIdx.x;
    if (t >= Gc * OUTc) return;
    const int g = t / OUTc, o = t - g * OUTc;
    float s = bc[o];
#pragma unroll 8
    for (int k = 0; k < HIDc; ++k)
        s += h[(size_t)g * HIDc + k] * Wc[(size_t)k * OUTc + o];
    out[t] = s;
}

extern "C" void kernel_launch(void* const* d_in, const int* in_sizes, int n_in,
                              void* d_out, int out_size, void* d_ws, size_t ws_size,
                              hipStream_t stream) {
    const float* x_in  = (const float*)d_in[0];
    const int*   ei    = (const int*)d_in[1];
    const int*   batch = (const int*)d_in[2];
    const float* Wq = (const float*)d_in[3];
    const float* bq = (const float*)d_in[4];
    const float* Wk = (const float*)d_in[5];
    const float* bk = (const float*)d_in[6];
    const float* Wv = (const float*)d_in[7];
    const float* bv = (const float*)d_in[8];
    const float* Ws = (const float*)d_in[9];
    const float* bs = (const float*)d_in[10];
    const float* W1 = (const float*)d_in[11];
    const float* b1 = (const float*)d_in[12];
    const float* W2 = (const float*)d_in[13];
    const float* b2 = (const float*)d_in[14];
    const float* Wc = (const float*)d_in[15];
    const float* bc = (const float*)d_in[16];
    float* out = (float*)d_out;

    // workspace layout (floats)
    const size_t NF = (size_t)Nn * HIDc;       // 2,560,000
    float* ws = (float*)d_ws;
    float* xQ = ws;
    float* xK = xQ + NF;
    float* xV = xK + NF;
    float* B0 = xV + NF;
    float* B1 = B0 + NF;
    float* logits   = B1 + NF;                 // E*8 floats
    unsigned* nodemax = (unsigned*)(logits + (size_t)Ee * HEADSc);
    float* denom = (float*)(nodemax + (size_t)Nn * HEADSc);
    float* pool  = denom + (size_t)Nn * HEADSc;
    float* h1    = pool + (size_t)Gc * HIDc;
    float* h2    = h1 + (size_t)Gc * HIDc;

    const int TB = 256;
    const int gemm_grid_N = Nn / 16;           // 1250 (N % 16 == 0)
    const int eh_grid  = (Ee * HEADSc + TB - 1) / TB;
    const int es_grid  = (Ee * 32 + TB - 1) / TB;
    const int nf_grid  = (int)((NF + TB - 1) / TB);

    const float* xcur = x_in;
    float* pingpong[2] = { B0, B1 };

    for (int l = 0; l < Lc; ++l) {
        float* xnext = pingpong[l & 1];
        const size_t wo = (size_t)l * HIDc * HIDc;
        const size_t bo = (size_t)l * HIDc;

        // projections (Q,K,V) + skip (written straight into the accumulator)
        gemm128_wmma<<<gemm_grid_N, TB, 0, stream>>>(xcur, Wq + wo, bq + bo, xQ, 0);
        gemm128_wmma<<<gemm_grid_N, TB, 0, stream>>>(xcur, Wk + wo, bk + bo, xK, 0);
        gemm128_wmma<<<gemm_grid_N, TB, 0, stream>>>(xcur, Wv + wo, bv + bo, xV, 0);
        gemm128_wmma<<<gemm_grid_N, TB, 0, stream>>>(xcur, Ws + wo, bs + bo, xnext, 0);

        // zero nodemax (ordered-uint min) and denom in one shot (contiguous)
        fill_zero_u32<<<(2 * Nn * HEADSc + TB - 1) / TB, TB, 0, stream>>>(
            nodemax, 2 * Nn * HEADSc);

        edge_logits<<<eh_grid, TB, 0, stream>>>(xQ, xK, ei, logits, nodemax);
        edge_exp<<<eh_grid, TB, 0, stream>>>(logits, nodemax, denom, ei);
        edge_scatter<<<es_grid, TB, 0, stream>>>(logits, denom, xV, ei, xnext);
        relu_ip<<<nf_grid, TB, 0, stream>>>(xnext, (int)NF);

        xcur = xnext;
    }

    // global add pool -> [G, HID]
    fill_zero_u32<<<(Gc * HIDc + TB - 1) / TB, TB, 0, stream>>>((unsigned*)pool, Gc * HIDc);
    pool_kernel<<<(Nn * HIDc + TB - 1) / TB, TB, 0, stream>>>(xcur, batch, pool);

    // MLP head (G=64 rows, multiple of 16) with fused ReLU
    gemm128_wmma<<<Gc / 16, TB, 0, stream>>>(pool, W1, b1, h1, 1);
    gemm128_wmma<<<Gc / 16, TB, 0, stream>>>(h1, W2, b2, h2, 1);

    classifier<<<(Gc * OUTc + TB - 1) / TB, TB, 0, stream>>>(h2, Wc, bc, out);
}